// MMCL_26912265077051
// MI455X (gfx1250) — compile-verified
//
#include <hip/hip_runtime.h>
#include <hip/hip_bf16.h>
#include <stdint.h>

// Problem constants (from reference)
#define M_ROWS 1024
#define N_COLS 65536
#define K_HARD 655            // int(0.01 * 65535)
#define BLOCK  1024           // 32 wave32 waves per workgroup

// In-LDS select tuning
#define PRIV    16            // privatized histogram copies (one per wave pair)
#define HSTRIDE 257           // padded stride (dwords): copy c, bin d -> bank (c+d)%64
#define CAP     8192          // candidate buffer capacity (guarded fallback if exceeded)

// dynamic LDS layout (dwords):
//   [0,            N_COLS)  s_data   : row values (256 KB)
//   [+PRIV*HSTRIDE        ) s_ph     : privatized 256-bin histograms (padded)
//   [+256                 ) s_hist   : reduced global histogram
//   [+CAP                 ) s_cand   : compacted candidate keys
#define SMEM_DWORDS (N_COLS + PRIV * HSTRIDE + 256 + CAP)

typedef unsigned int v4u __attribute__((ext_vector_type(4)));
typedef int          v8i __attribute__((ext_vector_type(8)));
typedef int          v4i __attribute__((ext_vector_type(4)));

// Monotonic float -> uint key (larger key == larger float)
__device__ __forceinline__ unsigned f2ord(float f) {
    unsigned b = __float_as_uint(f);
    return (b & 0x80000000u) ? ~b : (b | 0x80000000u);
}
__device__ __forceinline__ float ord2f(unsigned u) {
    unsigned b = (u & 0x80000000u) ? (u ^ 0x80000000u) : ~u;
    return __uint_as_float(b);
}

// One workgroup per row. TDM-DMA the full 256KB row into LDS (fits only in
// CDNA5's 320KB/WGP), then exact radix-select + scored scan entirely in LDS.
__global__ void __launch_bounds__(BLOCK)
hnm_row_kernel(const float* __restrict__ inputs,
               const int*   __restrict__ targets,
               float*       __restrict__ row_out) {
    extern __shared__ unsigned smem_u[];
    float*    s_data = (float*)smem_u;
    unsigned* s_ph   = smem_u + N_COLS;
    unsigned* s_hist = s_ph + PRIV * HSTRIDE;
    unsigned* s_cand = s_hist + 256;

    __shared__ unsigned s_sel[4];      // [0]=prefix, [1]=remaining, [2]=bin count, [3]=append counter
    __shared__ float    s_part[BLOCK / 32];

    const unsigned row = blockIdx.x;
    const unsigned tid = threadIdx.x;
    const unsigned tgt = (unsigned)targets[row];

    // ---- Phase 0: Tensor Data Mover: HBM row -> LDS (single 256KB DMA) ----
    // Row viewed as a 16384 x 4 tensor of 4-byte elements (tile_dim0 is 16-bit),
    // contiguous (tensor_dim0_stride = 16384) -> 65536 contiguous floats in LDS.
    if (tid < 32) {   // wave 0 issues one TDM op (EXEC ignored by TDM)
        const unsigned long long ga =
            (unsigned long long)(uintptr_t)inputs +
            (unsigned long long)row * (unsigned long long)(N_COLS * 4u);
        const unsigned lds = (unsigned)(uintptr_t)(&s_data[0]); // low 32b = LDS offset

        v4u g0;
        g0[0] = 1u;                          // count=1 (valid descriptor), user mode
        g0[1] = lds;                         // lds_addr (bytes)
        g0[2] = (unsigned)ga;                // global_addr[31:0]
        g0[3] = ((unsigned)(ga >> 32) & 0x01FFFFFFu) | 0x80000000u; // addr[56:32] | type=2

        v8i g1;
        g1[0] = 0x00020000;                  // data_size=2 (4 bytes); wg_mask=0
        g1[1] = 0x40000000;                  // tensor_dim0 = 16384 (bits[31:16])
        g1[2] = 0x00040000;                  // tensor_dim1 = 4     (bits[31:16])
        g1[3] = 0x40000000;                  // tile_dim0  = 16384  (bits[31:16])
        g1[4] = 4;                           // tile_dim1 = 4, tile_dim2 = 0
        g1[5] = 16384;                       // tensor_dim0_stride[31:0]
        g1[6] = 0;
        g1[7] = 0;

        v4i g2 = {0, 0, 0, 0};               // 2D tensor: groups 2/3 zero
        v4i g3 = {0, 0, 0, 0};
        v8i g4 = {0, 0, 0, 0, 0, 0, 0, 0};   // extra operand in clang-23 6-arg form

        __builtin_amdgcn_tensor_load_to_lds(g0, g1, g2, g3, g4, 0);
        __builtin_amdgcn_s_wait_tensorcnt(0);
    }

    // ---- Scan A: privatized top-byte histogram (bank-padded, 16 copies) ----
    for (unsigned i = tid; i < PRIV * HSTRIDE; i += BLOCK) s_ph[i] = 0u;
    __syncthreads();   // also makes the TDM'd row visible to all waves

    unsigned* myh = s_ph + (tid >> 6) * HSTRIDE;   // wave pair -> private copy
    for (unsigned j = tid; j < N_COLS; j += BLOCK) {
        if (j == tgt) continue;                    // positive masked out
        const unsigned u = f2ord(s_data[j]);
        atomicAdd(&myh[u >> 24], 1u);
    }
    __syncthreads();

    if (tid < 256) {                               // reduce 16 copies -> global hist
        unsigned c = 0;
        #pragma unroll
        for (int p = 0; p < PRIV; ++p) c += s_ph[p * HSTRIDE + tid];
        s_hist[tid] = c;
    }
    __syncthreads();

    if (tid == 0) {                                // select top-byte bin
        unsigned r = K_HARD;
        for (int d = 255; d >= 0; --d) {
            const unsigned c = s_hist[d];
            if (r <= c) { s_sel[0] = (unsigned)d; s_sel[1] = r; s_sel[2] = c; break; }
            r -= c;
        }
        s_sel[3] = 0u;                             // append counter for scan B
    }
    __syncthreads();

    const unsigned d0   = s_sel[0];
    const unsigned cnt0 = s_sel[2];
    unsigned prefix = d0;
    unsigned rem    = s_sel[1];

    if (cnt0 <= CAP) {
        // ---- Scan B: compact candidate keys (top byte == d0) ----
        for (unsigned j = tid; j < N_COLS; j += BLOCK) {
            if (j == tgt) continue;
            const unsigned u = f2ord(s_data[j]);
            if ((u >> 24) == d0) s_cand[atomicAdd(&s_sel[3], 1u)] = u;
        }
        __syncthreads();
        const unsigned cc = s_sel[3];              // == cnt0

        // refinement passes over the tiny candidate set
        for (int pass = 1; pass < 4; ++pass) {
            const int shift = 24 - 8 * pass;
            if (tid < 256) s_hist[tid] = 0u;
            __syncthreads();
            for (unsigned j = tid; j < cc; j += BLOCK) {
                const unsigned u = s_cand[j];
                if ((u >> (shift + 8)) == prefix)
                    atomicAdd(&s_hist[(u >> shift) & 255u], 1u);
            }
            __syncthreads();
            if (tid == 0) {
                unsigned r = rem, p = prefix;
                for (int d = 255; d >= 0; --d) {
                    const unsigned c = s_hist[d];
                    if (r <= c) { p = (p << 8) | (unsigned)d; break; }
                    r -= c;
                }
                s_sel[0] = p; s_sel[1] = r;
            }
            __syncthreads();
            prefix = s_sel[0];
            rem    = s_sel[1];
            __syncthreads();
        }
    } else {
        // ---- Fallback (adversarial data): full-scan refinement passes ----
        for (int pass = 1; pass < 4; ++pass) {
            const int shift = 24 - 8 * pass;
            if (tid < 256) s_hist[tid] = 0u;
            __syncthreads();
            for (unsigned j = tid; j < N_COLS; j += BLOCK) {
                if (j == tgt) continue;
                const unsigned u = f2ord(s_data[j]);
                if ((u >> (shift + 8)) == prefix)
                    atomicAdd(&s_hist[(u >> shift) & 255u], 1u);
            }
            __syncthreads();
            if (tid == 0) {
                unsigned r = rem, p = prefix;
                for (int d = 255; d >= 0; --d) {
                    const unsigned c = s_hist[d];
                    if (r <= c) { p = (p << 8) | (unsigned)d; break; }
                    r -= c;
                }
                s_sel[0] = p; s_sel[1] = r;
            }
            __syncthreads();
            prefix = s_sel[0];
            rem    = s_sel[1];
            __syncthreads();
        }
    }

    const unsigned T = prefix;    // exact key of the K-th largest negative
                                  // rem = #elements equal to T that are taken

    // ---- Scan C: deterministic fixed-order sum of (1+x)^2 over keys > T ----
    float sum = 0.0f;
    for (unsigned j = tid; j < N_COLS; j += BLOCK) {
        if (j == tgt) continue;
        const float x = s_data[j];
        if (f2ord(x) > T) {
            const float t = 1.0f + x;
            sum += t * t;
        }
    }
    // block tree reduction (fixed order -> deterministic)
    #pragma unroll
    for (int o = 16; o > 0; o >>= 1) sum += __shfl_down(sum, o, 32);
    if ((tid & 31u) == 0) s_part[tid >> 5] = sum;
    __syncthreads();
    if (tid < 32) {
        float r = s_part[tid];
        #pragma unroll
        for (int o = 16; o > 0; o >>= 1) r += __shfl_down(r, o, 32);
        if (tid == 0) {
            const float xT = ord2f(T);
            const float tt = 1.0f + xT;
            const float neg_total = r + (float)rem * tt * tt;   // ties at threshold
            const float pos = s_data[tgt];
            const float dp  = 1.0f - pos;
            row_out[row] = 5.0f * dp * dp + neg_total * (1.0f / (float)K_HARD);
        }
    }
}

// Deterministic fixed-order mean over the 1024 per-row results.
__global__ void __launch_bounds__(1024)
mean_kernel(const float* __restrict__ row_vals, float* __restrict__ out) {
    __shared__ float s_part[32];
    float v = row_vals[threadIdx.x];
    #pragma unroll
    for (int o = 16; o > 0; o >>= 1) v += __shfl_down(v, o, 32);
    if ((threadIdx.x & 31u) == 0) s_part[threadIdx.x >> 5] = v;
    __syncthreads();
    if (threadIdx.x < 32) {
        float r = s_part[threadIdx.x];
        #pragma unroll
        for (int o = 16; o > 0; o >>= 1) r += __shfl_down(r, o, 32);
        if (threadIdx.x == 0) out[0] = r * (1.0f / (float)M_ROWS);
    }
}

extern "C" void kernel_launch(void* const* d_in, const int* in_sizes, int n_in,
                              void* d_out, int out_size, void* d_ws, size_t ws_size,
                              hipStream_t stream) {
    const float* inputs  = (const float*)d_in[0];
    const int*   targets = (const int*)d_in[1];
    float* row_vals = (float*)d_ws;            // 1024 floats of scratch
    float* out      = (float*)d_out;

    hnm_row_kernel<<<M_ROWS, BLOCK, SMEM_DWORDS * sizeof(unsigned), stream>>>(inputs, targets, row_vals);
    mean_kernel<<<1, 1024, 0, stream>>>(row_vals, out);
}